// my_entropy_loss_66408784331217
// MI455X (gfx1250) — compile-verified
//
#include <hip/hip_runtime.h>
#include <hip/hip_bf16.h>

// Problem constants (match reference setup_inputs).
#define B_SAMPLES 128
#define N_PIX     196608
#define NBINS     256
#define SPLIT     8                       // blocks per sample
#define CHUNK     (N_PIX / SPLIT)         // 24576 floats per block
#define THREADS   256                     // 8 wave32 waves
#define NSUB      8                       // one sub-histogram per wave
#define TILES     (CHUNK / (THREADS * 4)) // 24 tiles of 256 float4

#if defined(__HIP_DEVICE_COMPILE__) && defined(__gfx1250__)
#define ASYNC_PATH 1
#else
#define ASYNC_PATH 0
#endif

#if ASYNC_PATH
typedef __attribute__((ext_vector_type(4))) int v4i_t;
typedef __attribute__((address_space(3))) char LAS3;

__device__ __forceinline__ void async_copy_b128(const void* g, void* l) {
#if __has_builtin(__builtin_amdgcn_global_load_async_to_lds_b128)
  // Probe-revealed prototype: (v4i*, v4i*, imm offset, imm cpol)
  __builtin_amdgcn_global_load_async_to_lds_b128((v4i_t*)g, (v4i_t*)l, 0, 0);
#else
  unsigned int lds_off = (unsigned int)(size_t)(LAS3*)l;
  asm volatile("global_load_async_to_lds_b128 %0, %1, off"
               :: "v"(lds_off), "v"(g) : "memory");
#endif
}

template <int N>
__device__ __forceinline__ void wait_asynccnt_le() {
#if __has_builtin(__builtin_amdgcn_s_wait_asynccnt)
  __builtin_amdgcn_s_wait_asynccnt(N);
#else
  asm volatile("s_wait_asynccnt %0" :: "i"(N) : "memory");
#endif
}
#endif // ASYNC_PATH

// ---------------------------------------------------------------------------
// Kernel 1: per-block partial histogram. Block (b*SPLIT + s) handles CHUNK
// contiguous floats of row b. LDS: 8 per-wave sub-histograms (ds_add_u32)
// + double-buffered async staging tiles (16 B per lane per tile).
// ---------------------------------------------------------------------------
__global__ __launch_bounds__(THREADS) void hist_kernel(
    const float* __restrict__ output, unsigned int* __restrict__ wsHist) {
  __shared__ unsigned int sub[NSUB * NBINS];   // 8 KB
#if ASYNC_PATH
  __shared__ float4 stage[2][THREADS];         // 8 KB double buffer
#endif
  const int tid  = threadIdx.x;
  const int wave = tid >> 5;                   // wave32: 8 waves per block
  unsigned int* myhist = &sub[(wave & (NSUB - 1)) * NBINS];

  for (int i = tid; i < NSUB * NBINS; i += THREADS) sub[i] = 0u;
  __syncthreads();

  const int blk = blockIdx.x;                  // == b*SPLIT + s, N = SPLIT*CHUNK
  const float4* __restrict__ src =
      (const float4*)output + (size_t)blk * (CHUNK / 4);

  auto bin1 = [&](float x) {
    int i = (int)(x * 256.0f);
    i = i < 0 ? 0 : (i > (NBINS - 1) ? (NBINS - 1) : i);
    atomicAdd(&myhist[i], 1u);                 // ds_add_u32
  };

#if ASYNC_PATH
  // Double-buffered async global->LDS pipeline: fetch tile t+1 while the
  // DS/VALU pipes histogram tile t.
  async_copy_b128(&src[tid], &stage[0][tid]);
  for (int t = 0; t < TILES; ++t) {
    if (t + 1 < TILES) {
      async_copy_b128(&src[(size_t)(t + 1) * THREADS + tid],
                      &stage[(t + 1) & 1][tid]);
      wait_asynccnt_le<1>();                   // oldest (tile t) landed in LDS
    } else {
      wait_asynccnt_le<0>();
    }
    float4 v = stage[t & 1][tid];              // ds_load_b128 (own slot only)
    bin1(v.x); bin1(v.y); bin1(v.z); bin1(v.w);
  }
#else
  for (int t = 0; t < TILES; ++t) {
    float4 v = src[(size_t)t * THREADS + tid]; // global_load_b128
    bin1(v.x); bin1(v.y); bin1(v.z); bin1(v.w);
  }
#endif

  __syncthreads();
  // Reduce 8 sub-histograms; direct store (no global atomics -> deterministic,
  // every slot fully overwritten each call, no ws zeroing needed).
  unsigned int total = 0;
  for (int w = 0; w < NSUB; ++w) total += sub[w * NBINS + tid];
  wsHist[(size_t)blk * NBINS + tid] = total;
}

// ---------------------------------------------------------------------------
// Kernel 2: per-sample entropy + squared error. Block b, thread t owns bin t.
// ---------------------------------------------------------------------------
__global__ __launch_bounds__(THREADS) void entropy_kernel(
    const unsigned int* __restrict__ wsHist, const float* __restrict__ target,
    float* __restrict__ wsSq) {
  __shared__ float red[THREADS];
  const int b = blockIdx.x, t = threadIdx.x;

  unsigned int c = 0;
  for (int s = 0; s < SPLIT; ++s)
    c += wsHist[((size_t)b * SPLIT + s) * NBINS + t];

  float term = 0.0f;
  if (c > 0u) {
    float p = (float)c * (1.0f / (float)N_PIX);  // p >= 1/N > 1e-12
    term = -p * __log2f(p);                      // v_log_f32
  }
  red[t] = term;
  __syncthreads();
  for (int off = THREADS / 2; off > 0; off >>= 1) {
    if (t < off) red[t] += red[t + off];
    __syncthreads();
  }
  if (t == 0) {
    float d = red[0] - target[b];
    wsSq[b] = d * d;
  }
}

// ---------------------------------------------------------------------------
// Kernel 3: mean over B squared errors -> d_out[0].
// ---------------------------------------------------------------------------
__global__ __launch_bounds__(B_SAMPLES) void mse_kernel(
    const float* __restrict__ wsSq, float* __restrict__ out) {
  __shared__ float red[B_SAMPLES];
  const int t = threadIdx.x;
  red[t] = wsSq[t];
  __syncthreads();
  for (int off = B_SAMPLES / 2; off > 0; off >>= 1) {
    if (t < off) red[t] += red[t + off];
    __syncthreads();
  }
  if (t == 0) out[0] = red[0] * (1.0f / (float)B_SAMPLES);
}

extern "C" void kernel_launch(void* const* d_in, const int* in_sizes, int n_in,
                              void* d_out, int out_size, void* d_ws, size_t ws_size,
                              hipStream_t stream) {
  (void)in_sizes; (void)n_in; (void)out_size; (void)ws_size;
  const float* output = (const float*)d_in[0];   // [B, N] float32
  const float* target = (const float*)d_in[1];   // [B]    float32
  float* out = (float*)d_out;                    // [1]    float32

  unsigned int* wsHist = (unsigned int*)d_ws;    // B*SPLIT*NBINS u32 = 1 MB
  float* wsSq = (float*)((char*)d_ws +
                         (size_t)B_SAMPLES * SPLIT * NBINS * sizeof(unsigned int));

  hist_kernel<<<B_SAMPLES * SPLIT, THREADS, 0, stream>>>(output, wsHist);
  entropy_kernel<<<B_SAMPLES, THREADS, 0, stream>>>(wsHist, target, wsSq);
  mse_kernel<<<1, B_SAMPLES, 0, stream>>>(wsSq, out);
}